// ESPmodule_73486890434670
// MI455X (gfx1250) — compile-verified
//
#include <hip/hip_runtime.h>
#include <math.h>

// Problem dims (fixed by reference: o is (8,1,512,512) f32)
constexpr int NB = 8;
constexpr int NH = 512;
constexpr int NW = 512;
constexpr int NPIX = NH * NW;                     // 262144
constexpr long long NTOT = (long long)NB * NPIX;  // 2097152
constexpr int ITERS = 50;

typedef float v8f __attribute__((ext_vector_type(8)));
typedef float v2f __attribute__((ext_vector_type(2)));

// Normalized 1-D gaussian taps (sigma=5, radius 2)
__device__ __forceinline__ float gband(int d) {
    float r = 0.0f;
    if (d == 0 || d == 4)      r = 0.19205064f;
    else if (d == 1 || d == 3) r = 0.20392638f;
    else if (d == 2)           r = 0.20804596f;
    return r;
}

__device__ __forceinline__ v8f wmma4(v2f a, v2f b, v8f c) {
    // V_WMMA_F32_16X16X4_F32 : D(16x16 f32) = A(16x4 f32) x B(4x16 f32) + C
    return __builtin_amdgcn_wmma_f32_16x16x4_f32(
        false, a, false, b, (short)0, c, false, false);
}

__device__ __forceinline__ unsigned lds_off(const void* p) {
    // generic pointer to LDS: low 32 bits are the LDS byte address
    return (unsigned)(size_t)p;
}

// ---------------------------------------------------------------------------
// Kernel 1: p = gauss5x5(1-2u) via f32 WMMA band-matmuls + raw moments of u.
// Block = 4 waves; wave = one 16x16 p tile. Input staged global->LDS with
// async b128 copies (ASYNCcnt), WMMA operands read from LDS.
// ---------------------------------------------------------------------------
__global__ __launch_bounds__(128)
void conv_moments_kernel(const float* __restrict__ u,
                         float* __restrict__ p,
                         float* __restrict__ mom)
{
    const int tid  = threadIdx.x;
    const int lane = tid & 31;
    const int wv   = tid >> 5;
    const int b    = blockIdx.z;
    const int row0 = blockIdx.y * 16;
    const int cb   = blockIdx.x * 64;          // block column base
    const float* ub = u + (size_t)b * NPIX;

    const int n     = lane & 15;               // N index / A-layout M index
    const int kbase = (lane < 16) ? 0 : 2;     // K base per half-wave
    const int Mrow  = (lane < 16) ? 0 : 8;     // C-layout row offset

    __shared__ float tIn[20 * 84];             // staged u: rows row0-2.., cols cb-4.. (80 used, pitch 84)
    __shared__ float outv[4][16 * 22];         // per-wave 16x20 vertical result (pitch 22)

    // ---- async stage: 20 rows x 80 cols = 400 b128 transfers, clamped addrs ----
    for (int i = tid; i < 400; i += 128) {
        const int rr = i / 20;                 // staged row 0..19
        const int sg = i % 20;                 // 4-float segment
        int gr = row0 - 2 + rr;
        int gc = cb - 4 + sg * 4;
        gr = min(max(gr, 0), NH - 1);
        gc = min(max(gc, 0), NW - 4);
        const float* src = ub + gr * NW + gc;
        const unsigned dst = lds_off(&tIn[rr * 84 + sg * 4]);
        asm volatile("global_load_async_to_lds_b128 %0, %1, off"
                     :: "v"(dst), "v"((unsigned long long)(size_t)src)
                     : "memory");
    }
    asm volatile("s_wait_asynccnt 0x0" ::: "memory");
    __syncthreads();

    // ---- zero-pad fixup, only for image-edge blocks (uniform branch) ----
    const bool edge = (row0 == 0) | (row0 == NH - 16) | (cb == 0) | (cb == NW - 64);
    if (edge) {
        for (int i = tid; i < 20 * 80; i += 128) {
            const int rr = i / 80, jj = i % 80;
            const int gr = row0 - 2 + rr, gc = cb - 4 + jj;
            if (gr < 0 || gr >= NH || gc < 0 || gc >= NW)
                tIn[rr * 84 + jj] = 0.5f;      // u=0.5 => (1-2u)=0 (zero pad)
        }
        __syncthreads();
    }

    // ---- constant A operand: vertical band matrix G[i][j]=g[j-i], K chunks of 4 ----
    v2f aG[5];
#pragma unroll
    for (int c = 0; c < 5; ++c) {
        aG[c].x = gband(4 * c + kbase     - n);
        aG[c].y = gband(4 * c + kbase + 1 - n);
    }

    // ---- vertical pass: two WMMA tiles cover the 20-col halo for this wave ----
#pragma unroll
    for (int t = 0; t < 2; ++t) {
        const int j0 = wv * 16 + (t ? 16 : 0) + n;   // staged col of lane's B column
        v8f acc = {};
#pragma unroll
        for (int c = 0; c < 5; ++c) {
            const int rr = 4 * c + kbase;
            v2f bv;
            bv.x = 1.0f - 2.0f * tIn[ rr      * 84 + j0];
            bv.y = 1.0f - 2.0f * tIn[(rr + 1) * 84 + j0];
            acc = wmma4(aG[c], bv, acc);
        }
        const int oc    = t ? (14 + n) : (n - 2);    // outv col j = col-(col0-2)
        const bool ocok = t ? (n <= 5) : (n >= 2);
        if (ocok) {
#pragma unroll
            for (int r = 0; r < 8; ++r)
                outv[wv][(r + Mrow) * 22 + oc] = acc[r];
        }
    }
    __syncthreads();

    // ---- horizontal pass: P = outv(16x20) x Gh(20x16), Gh[j][x]=g[j-x] ----
    v8f pc = {};
#pragma unroll
    for (int c = 0; c < 5; ++c) {
        v2f av = *(const v2f*)&outv[wv][n * 22 + 4 * c + kbase]; // ds_load_b64, 8B aligned
        v2f bh;
        bh.x = gband(4 * c + kbase     - n);
        bh.y = gband(4 * c + kbase + 1 - n);
        pc = wmma4(av, bh, pc);
    }
    {
        float* pb = p + (size_t)b * NPIX;
#pragma unroll
        for (int r = 0; r < 8; ++r)
            pb[(row0 + r + Mrow) * NW + (cb + wv * 16 + n)] = pc[r];
    }

    // ---- raw moments of u over this wave's core 16x16 (from staged LDS) ----
    float s = 0.f, sx = 0.f, sy = 0.f, sxx = 0.f, syy = 0.f, sxy = 0.f;
    {
        const int coff = 4 + wv * 16;                 // staged col of core col0
        const float Y = (float)(cb + wv * 16 + n + 1);
#pragma unroll
        for (int r = 0; r < 8; ++r) {
            const float uv = tIn[(2 + r + Mrow) * 84 + coff + n];
            const float X = (float)(row0 + r + Mrow + 1);
            s  += uv;      sx  += uv * X;     sy  += uv * Y;
            sxx += uv * X * X; syy += uv * Y * Y; sxy += uv * X * Y;
        }
    }
#pragma unroll
    for (int off = 16; off > 0; off >>= 1) {
        s   += __shfl_xor(s,   off, 32);
        sx  += __shfl_xor(sx,  off, 32);
        sy  += __shfl_xor(sy,  off, 32);
        sxx += __shfl_xor(sxx, off, 32);
        syy += __shfl_xor(syy, off, 32);
        sxy += __shfl_xor(sxy, off, 32);
    }
    if (lane == 0) {
        float* m = mom + b * 6;
        atomicAdd(m + 0, s);   atomicAdd(m + 1, sx);  atomicAdd(m + 2, sy);
        atomicAdd(m + 3, sxx); atomicAdd(m + 4, syy); atomicAdd(m + 5, sxy);
    }
}

// ---------------------------------------------------------------------------
// Kernel 2: q_new, Tq (backward-div of Tx*q_new, Ty*q_new), u_new / final out.
// Branch-free: all neighbor loads issue up-front with clamped indices,
// boundary handling via selects (v_cndmask), so loads pipeline.
// ---------------------------------------------------------------------------
__global__ __launch_bounds__(256)
void update_kernel(const float* __restrict__ o,
                   const float* __restrict__ uin,
                   const float* __restrict__ qin,
                   const float* __restrict__ p,
                   const float* __restrict__ mom,
                   float* __restrict__ uout,
                   float* __restrict__ qout,
                   float* __restrict__ out,
                   int last)
{
    const int b  = blockIdx.z;
    const int tx = threadIdx.x & 63;
    const int ty = threadIdx.x >> 6;
    const int c  = blockIdx.x * 64 + tx;
    const int r  = blockIdx.y * 4 + ty;
    const size_t idx = (size_t)b * NPIX + r * NW + c;

    const float* ub = uin + (size_t)b * NPIX;
    const float* qb = qin + (size_t)b * NPIX;

    const int rm = max(r - 1, 0), rp = min(r + 1, NH - 1);
    const int cm = max(c - 1, 0), cp = min(c + 1, NW - 1);
    const bool okU = (r > 0), okL = (c > 0);
    const bool okD = (r + 1 < NH), okR = (c + 1 < NW);

    // issue all loads unconditionally (clamped, in-bounds)
    float u_cc = ub[r  * NW + c ];
    float u_dc = ub[rp * NW + c ];
    float u_cr = ub[r  * NW + cp];
    float u_uc = ub[rm * NW + c ];
    float u_ur = ub[rm * NW + cp];
    float u_cl = ub[r  * NW + cm];
    float u_dl = ub[rp * NW + cm];
    const float q_cc = qb[r  * NW + c ];
    const float q_uc = qb[rm * NW + c ];
    const float q_cl = qb[r  * NW + cm];
    const float ov = o[idx];
    const float pv = p[idx];

    const float* m = mom + b * 6;
    const float s   = m[0];
    const float cx  = m[1] / s;
    const float cy  = m[2] / s;
    const float css = m[5] / s - cx * cy;   // cossin
    const float bsn = m[3] / s - cx * cx;   // bsin
    const float asn = m[4] / s - cy * cy;   // asin

    // zero-pad selects
    u_dc = okD ? u_dc : 0.0f;
    u_cr = okR ? u_cr : 0.0f;
    u_ur = okR ? u_ur : 0.0f;
    u_dl = okD ? u_dl : 0.0f;

    auto TxTy = [&](int rr, int ccn, float& Tx, float& Ty) {
        float dx = (float)(rr + 1) - cx;        // reference x = row coord
        float dy = (float)(ccn + 1) - cy;       // reference y = col coord
        float txv = -dx * css + dy * bsn;
        float tyv =  dy * css - dx * asn;
        float tn = sqrtf(txv * txv + tyv * tyv) + 1e-10f;
        Tx = txv / tn; Ty = tyv / tn;
    };

    float Txc, Tyc; TxTy(r,     c,     Txc, Tyc);
    float Txu, Tyu; TxTy(r - 1, c,     Txu, Tyu);
    float Txl, Tyl; TxTy(r,     c - 1, Txl, Tyl);

    // q_new at center and the two causal neighbors (TAU = 1)
    const float qc = q_cc - ((u_dc - u_cc) * Txc + (u_cr - u_cc) * Tyc);
    const float qu = q_uc - ((u_cc - u_uc) * Txu + (u_ur - u_uc) * Tyu);
    const float ql = q_cl - ((u_dl - u_cl) * Txl + (u_cc - u_cl) * Tyl);

    // Tq(r,c) = Txq(r,c)-Txq(r-1,c) + Tyq(r,c)-Tyq(r,c-1), zero-padded
    float div = (Txc + Tyc) * qc;
    div -= okU ? Txu * qu : 0.0f;
    div -= okL ? Tyl * ql : 0.0f;

    const float z = ov - pv - div;              // LAM = EPS = 1
    qout[idx] = qc;
    if (last) out[idx]  = z;                    // u1 = (o - p - Tq)/EPS
    else      uout[idx] = 1.0f / (1.0f + expf(-z));
}

// ---------------------------------------------------------------------------
// Init: u0 = sigmoid(o), q0 = 0, zero all per-iteration moment slots.
// ---------------------------------------------------------------------------
__global__ void init_kernel(const float* __restrict__ o,
                            float* __restrict__ u0,
                            float* __restrict__ q0,
                            float* __restrict__ mom, int momN)
{
    long long i = (long long)blockIdx.x * blockDim.x + threadIdx.x;
    if (i < NTOT) {
        float z = o[i];
        u0[i] = 1.0f / (1.0f + expf(-z));
        q0[i] = 0.0f;
    }
    if (i < momN) mom[i] = 0.0f;
}

extern "C" void kernel_launch(void* const* d_in, const int* in_sizes, int n_in,
                              void* d_out, int out_size, void* d_ws, size_t ws_size,
                              hipStream_t stream)
{
    (void)in_sizes; (void)n_in; (void)out_size; (void)ws_size;
    const float* o = (const float*)d_in[0];
    float* out = (float*)d_out;
    float* ws  = (float*)d_ws;

    // scratch layout (floats): 5 image stacks + per-iteration moment slots
    float* uA  = ws;
    float* uB  = ws + 1 * NTOT;
    float* qA  = ws + 2 * NTOT;
    float* qB  = ws + 3 * NTOT;
    float* pB  = ws + 4 * NTOT;
    float* mom = ws + 5 * NTOT;                 // ITERS * NB * 6 floats
    const int momN = ITERS * NB * 6;

    init_kernel<<<(int)((NTOT + 255) / 256), 256, 0, stream>>>(o, uA, qA, mom, momN);

    dim3 g1(NW / 64, NH / 16, NB);   // conv+moments: 8 x 32 x 8 blocks of 128
    dim3 g2(NW / 64, NH / 4,  NB);   // update:       8 x 128 x 8 blocks of 256
    for (int it = 0; it < ITERS; ++it) {
        float* uin  = (it & 1) ? uB : uA;
        float* uout = (it & 1) ? uA : uB;
        float* qin  = (it & 1) ? qB : qA;
        float* qout = (it & 1) ? qA : qB;
        float* mslot = mom + (size_t)it * NB * 6;
        conv_moments_kernel<<<g1, 128, 0, stream>>>(uin, pB, mslot);
        update_kernel<<<g2, 256, 0, stream>>>(o, uin, qin, pB, mslot,
                                              uout, qout, out,
                                              (it == ITERS - 1) ? 1 : 0);
    }
}